// AttentionFlow_8985071583852
// MI455X (gfx1250) — compile-verified
//
#include <hip/hip_runtime.h>
#include <hip/hip_bf16.h>

#define B_     8
#define N_     50000
#define D_     32
#define N_SRC_ 12500
#define DEG_   8
#define R_     500
#define E_B_   (N_SRC_ * DEG_)
#define E_     (B_ * E_B_)       /* 800000 edges  */
#define NA_    (B_ * N_)         /* 400000 floats */

#define NBLK_  1024              /* persistent blocks */

typedef __attribute__((ext_vector_type(4))) unsigned int v4u;
typedef __attribute__((ext_vector_type(8))) int v8i;
typedef __attribute__((ext_vector_type(4))) int v4i;

__global__ void zero_kernel(float* __restrict__ p, int n) {
  int i = blockIdx.x * blockDim.x + threadIdx.x;
  if (i < n) p[i] = 0.0f;
}

__device__ __forceinline__ float edge_term(float a, float x, float u, float r,
                                           float w0, float w1, float w2, float w3,
                                           float bb, float ow) {
  // h = a*x*w0 + a*r*x*w1 + a*u*w2 + a*r*u*w3 + b
  //   = a*( x*(w0 + r*w1) + u*(w2 + r*w3) ) + b ; then relu(h)*ow  (out_b summed once)
  float h = a * (x * fmaf(r, w1, w0) + u * fmaf(r, w3, w2)) + bb;
  h = fmaxf(h, 0.0f);
  return h * ow;
}

__launch_bounds__(256)
__global__ void attnflow_kernel(const float* __restrict__ node_attention,  // (B,N)
                                const float* __restrict__ hidden_con,      // (B,N,D)
                                const float* __restrict__ hidden_uncon,    // (1,N,D)
                                const float* __restrict__ edges_y,         // (E,)
                                const float* __restrict__ rel_emb,         // (R,D)
                                const float* __restrict__ ws,              // (4,D)
                                const float* __restrict__ b,               // (D,)
                                const float* __restrict__ out_w,           // (D,)
                                const float* __restrict__ out_b,           // (D,)
                                const int*   __restrict__ se,              // (E,6)
                                float* __restrict__ trans_out,             // (E,)
                                float* __restrict__ na_out) {              // (B,N), pre-zeroed
  __shared__ __align__(16) float s_rel[R_ * D_];   // 64000 B: rel_emb table in LDS
  __shared__ __align__(16) float s_c[224];         // ws(128) | b(32) | out_w(32) | out_b(32)

  const int t = threadIdx.x;

  // Stage small const tables to LDS.
  if      (t < 128) s_c[t] = ws[t];
  else if (t < 160) s_c[t] = b[t - 128];
  else if (t < 192) s_c[t] = out_w[t - 160];
  else if (t < 224) s_c[t] = out_b[t - 192];

#if __has_builtin(__builtin_amdgcn_tensor_load_to_lds)
  // Tensor Data Mover: DMA the 500x32 f32 rel_emb table (64 KB) into LDS.
  // One D# per block, issued by wave 0; TENSORcnt drained before the WG barrier.
  if (t < 32) {
    unsigned long long ga = (unsigned long long)(uintptr_t)rel_emb;
    unsigned int lds_off = (unsigned int)(uintptr_t)&s_rel[0];  // low 32 bits = LDS byte offset
    v4u g0 = { 1u,                                                    // count=1, user D#
               lds_off,                                               // lds_addr
               (unsigned int)(ga & 0xFFFFFFFFu),                      // global_addr[31:0]
               (unsigned int)(((ga >> 32) & 0x01FFFFFFu) | (2u << 30)) }; // ga[56:32] | type=2
    v8i g1 = { (int)(2u << 16),          // data_size = 4 bytes
               (int)(D_ << 16),          // tensor_dim0 = 32
               (int)(R_ << 16),          // tensor_dim1 = 500
               (int)(D_ << 16),          // tile_dim0   = 32
               (int)R_,                  // tile_dim1   = 500, tile_dim2 = 0
               (int)D_,                  // tensor_dim0_stride = 32
               0, 0 };
    v4i gz = { 0, 0, 0, 0 };
#if __clang_major__ >= 23
    v8i gz8 = { 0, 0, 0, 0, 0, 0, 0, 0 };
    __builtin_amdgcn_tensor_load_to_lds(g0, g1, gz, gz, gz8, 0);
#else
    __builtin_amdgcn_tensor_load_to_lds(g0, g1, gz, gz, 0);
#endif
    __builtin_amdgcn_s_wait_tensorcnt(0);
  }
#else
  for (int i = t; i < R_ * D_; i += blockDim.x) s_rel[i] = rel_emb[i];
#endif
  __syncthreads();

  const float4* sc4 = (const float4*)s_c;
  const float4* sr4 = (const float4*)s_rel;

  // sum(out_b) is edge-invariant: fold it once per thread (saves 8 LDS float4/edge).
  float sum_ob = 0.0f;
#pragma unroll
  for (int c = 0; c < D_ / 4; ++c) {
    float4 ob = sc4[48 + c];
    sum_ob += ob.x + ob.y + ob.z + ob.w;
  }

  const long long stride = (long long)gridDim.x * blockDim.x;
  const long long base   = (long long)blockIdx.x * blockDim.x + t;

  // One thread per edge; 8-edge softmax segments align with 8-lane wave32 sub-groups
  // (E and stride are multiples of 8, so groups are never split).
  for (long long eid = base; eid < E_; eid += stride) {
    // Prefetch the next grid-stride edge record (global_prefetch_b8).
    if (eid + stride < E_) __builtin_prefetch(se + (eid + stride) * 6, 0, 3);

    int2 e01 = *(const int2*)(se + eid * 6);      // idx, vi   (24B stride => 8B aligned)
    int2 e23 = *(const int2*)(se + eid * 6 + 2);  // vj, rel
    const int idx = e01.x, vi = e01.y, vj = e23.x, rel = e23.y;

    const float4* avi = (const float4*)(hidden_con + ((long long)idx * N_ + vi) * D_);
    const float4* avj = (const float4*)(hidden_con + ((long long)idx * N_ + vj) * D_);
    const float4* auj = (const float4*)(hidden_uncon + (long long)vj * D_);
    const float4* rej = sr4 + rel * (D_ / 4);

    float logit = sum_ob;
#pragma unroll
    for (int c = 0; c < D_ / 4; ++c) {
      float4 a = avi[c], x = avj[c], u = auj[c], r = rej[c];
      float4 w0 = sc4[c],      w1 = sc4[8 + c], w2 = sc4[16 + c], w3 = sc4[24 + c];
      float4 bb = sc4[32 + c], ow = sc4[40 + c];
      logit += edge_term(a.x, x.x, u.x, r.x, w0.x, w1.x, w2.x, w3.x, bb.x, ow.x);
      logit += edge_term(a.y, x.y, u.y, r.y, w0.y, w1.y, w2.y, w3.y, bb.y, ow.y);
      logit += edge_term(a.z, x.z, u.z, r.z, w0.z, w1.z, w2.z, w3.z, bb.z, ow.z);
      logit += edge_term(a.w, x.w, u.w, r.w, w0.w, w1.w, w2.w, w3.w, bb.w, ow.w);
    }

    // Segment softmax over the 8-edge group via width-8 shfl butterflies.
    float m = logit;
    #pragma unroll
    for (int o = 1; o < 8; o <<= 1) m = fmaxf(m, __shfl_xor(m, o, 8));
    float ex = __expf(logit - m);
    float s = ex;
    #pragma unroll
    for (int o = 1; o < 8; o <<= 1) s += __shfl_xor(s, o, 8);
    float trans = ex / s;

    float ta = node_attention[(long long)idx * N_ + vi] * trans * edges_y[eid];
    trans_out[eid] = ta;
    atomicAdd(na_out + (long long)idx * N_ + vj, ta);
  }
}

extern "C" void kernel_launch(void* const* d_in, const int* in_sizes, int n_in,
                              void* d_out, int out_size, void* d_ws, size_t ws_size,
                              hipStream_t stream) {
  const float* na = (const float*)d_in[0];
  const float* hc = (const float*)d_in[1];
  const float* hu = (const float*)d_in[2];
  const float* ey = (const float*)d_in[3];
  const float* re = (const float*)d_in[4];
  const float* w  = (const float*)d_in[5];
  const float* b  = (const float*)d_in[6];
  const float* ow = (const float*)d_in[7];
  const float* ob = (const float*)d_in[8];
  const int*   se = (const int*)d_in[9];

  float* trans_out = (float*)d_out;        // first E_ floats
  float* na_out    = (float*)d_out + E_;   // next B*N floats (scatter-add target)

  zero_kernel<<<(NA_ + 255) / 256, 256, 0, stream>>>(na_out, NA_);
  attnflow_kernel<<<NBLK_, 256, 0, stream>>>(na, hc, hu, ey, re, w, b, ow, ob, se,
                                             trans_out, na_out);
}